// DiscreteEmbedding_15032385536232
// MI455X (gfx1250) — compile-verified
//
#include <hip/hip_runtime.h>
#include <stdint.h>

// DiscreteEmbedding: out[b,s,:] = sum_n rect_window(x[b,s]*N - n) * emb[n,:]
// rect_window is 4 inside (|t|<0.5), 1 exactly at |t|=0.5, else 0 -> a scaled
// one-hot gather, NOT a real matmul. Roofline: 16.8 MB of stores @ 23.3 TB/s
// ~= 0.75 us floor; WMMA (f32 16x16x4) would cost 4.3 GFLOP of dense work and
// be >50x slower, so the optimized MI455X kernel is a gather.
// CDNA5 paths used: async global->LDS staging (ASYNCcnt + s_wait_asynccnt) of
// the full 128 KB table into 320 KB/WGP LDS; ds_load_b128 gathers; b128
// non-temporal streaming stores.

#define NPOINTS 512
#define DEMB    64
#define WAVES_PER_BLOCK 8
#define THREADS (WAVES_PER_BLOCK * 32)
#define NBLOCKS 256
#define LDS_BYTES (NPOINTS * DEMB * sizeof(float))   // 131072 bytes

typedef __attribute__((ext_vector_type(4))) float f32x4;

__device__ __forceinline__ float sgnf(float v) {
    return (float)((v > 0.0f) - (v < 0.0f));
}

// Faithful to the reference: (sign(t+h)-sign(t-h))/2 * (1+sign(t+h)) * (1-sign(t-h))
__device__ __forceinline__ float rectw(float t) {
    float sp = sgnf(t + 0.5f);
    float sm = sgnf(t - 0.5f);
    return (sp - sm) * 0.5f * (1.0f + sp) * (1.0f - sm);
}

__global__ void __launch_bounds__(THREADS)
discrete_embedding_kernel(const float* __restrict__ x,
                          const float* __restrict__ emb,
                          float* __restrict__ out,
                          int nrows) {
    extern __shared__ f32x4 semb[];   // [NPOINTS][DEMB/4]
    const unsigned tid = threadIdx.x;

    // ---- Stage the whole embedding table into LDS via CDNA5 async copies ----
    {
        const uint64_t gbase = (uint64_t)(uintptr_t)emb;
        // Generic LDS pointer: low 32 bits are the LDS byte offset.
        const uint32_t lbase = (uint32_t)(uintptr_t)semb;
        const int chunks = (int)(LDS_BYTES / 16);      // 8192 x b128 transfers
        for (int i = (int)tid; i < chunks; i += THREADS) {
            uint32_t off = (uint32_t)i * 16u;
            uint64_t ga  = gbase + off;
            uint32_t la  = lbase + off;
            asm volatile("global_load_async_to_lds_b128 %0, %1, off"
                         :: "v"(la), "v"(ga) : "memory");
        }
#if __has_builtin(__builtin_amdgcn_s_wait_asynccnt)
        __builtin_amdgcn_s_wait_asynccnt(0);
#else
        asm volatile("s_wait_asynccnt 0" ::: "memory");
#endif
    }
    __syncthreads();

    // ---- Two rows per wave iteration: half-wave (16 lanes) x f32x4 = 64 cols.
    // Each lane does one ds_load_b128 gather and one global_store_b128.
    const int lane   = (int)(tid & 31u);
    const int half   = lane >> 4;          // 0: row r, 1: row r+1
    const int sub    = lane & 15;          // 4-column group within the row
    const int wave   = (int)(tid >> 5u);
    const int gw     = (int)blockIdx.x * WAVES_PER_BLOCK + wave;
    const int stride = (int)gridDim.x * WAVES_PER_BLOCK;
    const int npairs = (nrows + 1) >> 1;

    for (int p = gw; p < npairs; p += stride) {
        const int row = 2 * p + half;
        if (row >= nrows) continue;

        float xs = x[row] * (float)NPOINTS;
        float fn = floorf(xs);
        int   n0 = (int)fn;
        // Only floor(xs) and floor(xs)+1 can fall inside the width-1 window.
        float w0 = rectw(xs - fn);
        float w1 = rectw(xs - (fn + 1.0f));

        f32x4 acc;
        acc.x = 0.0f; acc.y = 0.0f; acc.z = 0.0f; acc.w = 0.0f;
        if (w0 != 0.0f && n0 < NPOINTS) {
            f32x4 e = semb[n0 * (DEMB / 4) + sub];
            acc.x += w0 * e.x; acc.y += w0 * e.y;
            acc.z += w0 * e.z; acc.w += w0 * e.w;
        }
        int n1 = n0 + 1;
        if (w1 != 0.0f && n1 < NPOINTS) {
            f32x4 e = semb[n1 * (DEMB / 4) + sub];
            acc.x += w1 * e.x; acc.y += w1 * e.y;
            acc.z += w1 * e.z; acc.w += w1 * e.w;
        }
        // Streaming store: 512 B contiguous per wave (2 rows), non-temporal.
        __builtin_nontemporal_store(
            acc, (f32x4*)(out + (size_t)row * DEMB + (size_t)sub * 4));
    }
}

extern "C" void kernel_launch(void* const* d_in, const int* in_sizes, int n_in,
                              void* d_out, int out_size, void* d_ws, size_t ws_size,
                              hipStream_t stream) {
    const float* x   = (const float*)d_in[0];   // [B,S] float32
    const float* emb = (const float*)d_in[1];   // [N,D] float32
    // d_in[2] (points) is arange(N); its values are folded into the window math.
    float* out = (float*)d_out;                  // [B,S,D] float32
    const int nrows = in_sizes[0];               // B*S = 65536

    discrete_embedding_kernel<<<NBLOCKS, THREADS, LDS_BYTES, stream>>>(
        x, emb, out, nrows);
}